// MixtureOfAgents_23270132810289
// MI455X (gfx1250) — compile-verified
//
#include <hip/hip_runtime.h>
#include <stdint.h>

// Problem constants (from the reference)
#define NTOK   16384          // B*T = 4*4096
#define CDIM   768
#define FFN    2048
#define EPA    2
#define NAG    10
#define NEXP   20

// Tiling
#define MT     32             // tokens per block
#define FC     32             // FFN chunk (stage-1 N / stage-2 K)
#define KC     64             // C chunk for stage-1 K loop
#define LDX    772            // x tile LDS row stride   (768+4, bank-friendly, 16B aligned)
#define LD1    68             // w1/w3 chunk row stride  (64+4)
#define LDH    36             // h / w2 chunk row stride (32+4)

// LDS layout (float offsets) -- w1/w3 chunk buffers are double-buffered
#define OFF_XS    0
#define OFF_W1S   (OFF_XS  + MT*LDX)        // 24704
#define OFF_W3S   (OFF_W1S + 2*FC*LD1)      // +4352
#define OFF_H1    (OFF_W3S + 2*FC*LD1)      // +4352
#define OFF_H3    (OFF_H1  + MT*LDH)        // +1152
#define OFF_HC    (OFF_H3  + MT*LDH)        // +1152
#define OFF_W2S   (OFF_HC  + MT*LDH)        // +1152
#define OFF_TOPW  (OFF_W2S + CDIM*LDH)      // +27648
#define OFF_TOPI  (OFF_TOPW + MT*2)
#define OFF_AEW   (OFF_TOPI + MT*2)
#define SMEM_FLOATS (OFF_AEW + 4*MT)        // 64768 floats = 259,072 B (< 320KB WGP LDS)

typedef float v2f __attribute__((ext_vector_type(2)));
typedef float v8f __attribute__((ext_vector_type(8)));
typedef int   v4i __attribute__((vector_size(16)));

static __device__ inline v8f wmma4(v2f a, v2f b, v8f c) {
  // D = A(16x4, f32) * B(4x16, f32) + C(16x16, f32), full fp32 precision
  return __builtin_amdgcn_wmma_f32_16x16x4_f32(false, a, false, b, (short)0, c, false, false);
}

// ---- async global->LDS copy (CDNA5 ASYNCcnt path), with safe fallback ----
#if __has_builtin(__builtin_amdgcn_global_load_async_to_lds_b128)
#define HAS_ASYNC 1
#else
#define HAS_ASYNC 0
#endif

static __device__ inline void cp16(float* l, const float* g) {
#if HAS_ASYNC
  typedef __attribute__((address_space(1))) v4i* gas_t;
  typedef __attribute__((address_space(3))) v4i* las_t;
  __builtin_amdgcn_global_load_async_to_lds_b128(
      (gas_t)(uintptr_t)g, (las_t)(uint32_t)(uintptr_t)l, 0, 0);
#else
  *(float4*)l = *(const float4*)g;
#endif
}

static __device__ inline void async_wait0() {
#if HAS_ASYNC
#if __has_builtin(__builtin_amdgcn_s_wait_asynccnt)
  __builtin_amdgcn_s_wait_asynccnt(0);
#else
  asm volatile("s_wait_asynccnt 0x0" ::: "memory");
#endif
#endif
}

// ---------------------------------------------------------------------------
// Kernel 1: gating. One wave per token.
//  - agent logits (10) -> softmax -> top2 -> renormalized weights + indices
//  - all 20 expert logits stored raw (ae softmax over a pair happens later)
//  - token 4095 (= (b=0, t=T-1)) publishes sel[0..1]
// ---------------------------------------------------------------------------
__global__ __launch_bounds__(256) void moa_gate(
    const float* __restrict__ x, const float* __restrict__ agw,
    const float* __restrict__ egw, int* __restrict__ sel,
    float* __restrict__ topw, int* __restrict__ topi, float* __restrict__ elog)
{
  const int tid  = threadIdx.x;
  const int lane = tid & 31;
  const int wv   = tid >> 5;
  const int64_t tok = (int64_t)blockIdx.x * 8 + wv;
  const float* xp = x + tok * CDIM;

  float xr[CDIM / 32];
#pragma unroll
  for (int j = 0; j < CDIM / 32; ++j) xr[j] = xp[lane + 32 * j];

  __shared__ float lg[8][32];

  for (int g = 0; g < NAG + NEXP; ++g) {
    const float* w = (g < NAG) ? (agw + (int64_t)g * CDIM)
                               : (egw + (int64_t)(g - NAG) * CDIM);
    float p = 0.f;
#pragma unroll
    for (int j = 0; j < CDIM / 32; ++j) p += xr[j] * w[lane + 32 * j];
#pragma unroll
    for (int off = 16; off; off >>= 1) p += __shfl_xor(p, off, 32);
    if (lane == 0) lg[wv][g] = p;
  }
  __syncthreads();

  if (lane == 0) {
    const float* l = lg[wv];
    float m = l[0];
    for (int i = 1; i < NAG; ++i) m = fmaxf(m, l[i]);
    float ev[NAG], s = 0.f;
    for (int i = 0; i < NAG; ++i) { ev[i] = __expf(l[i] - m); s += ev[i]; }
    const float inv = 1.f / s;
    int i0 = 0; float b0 = -1.f;
    for (int i = 0; i < NAG; ++i) { float w = ev[i] * inv; if (w > b0) { b0 = w; i0 = i; } }
    int i1 = 0; float b1 = -1.f;
    for (int i = 0; i < NAG; ++i) {
      if (i == i0) continue;
      float w = ev[i] * inv; if (w > b1) { b1 = w; i1 = i; }
    }
    const float nrm = 1.f / (b0 + b1 + 1e-6f);
    topw[tok * 2 + 0] = b0 * nrm;  topw[tok * 2 + 1] = b1 * nrm;
    topi[tok * 2 + 0] = i0;        topi[tok * 2 + 1] = i1;
    if (tok == 4095) { sel[0] = i0; sel[1] = i1; }   // top_i[0, -1, k]
  }
  if (lane < NEXP) elog[tok * NEXP + lane] = lg[wv][NAG + lane];
}

// ---------------------------------------------------------------------------
// Kernel 2: fused gated-FFN, 32-token tile per block, 256 threads = 8 waves.
//  out[t,c] = sum_k  aw[t,k]*(1+0.1*role_emb[idx[t,k],c]) *
//             sum_e  ae[t,e] * ( silu(x w1e^T) * (x w3e^T) ) w2e[c,:]
// Stage 1: waves 0-3 build the 4 16x16 tiles of x*w1^T, waves 4-7 of x*w3^T,
//          with double-buffered async w1/w3 chunk staging.
// Stage 2: all 8 waves update the 32x768 fp32 accumulator (6 N-tiles each);
//          the w2 chunk is async-issued a whole stage ahead of its use.
// ---------------------------------------------------------------------------
__global__ __launch_bounds__(256, 1) void moa_main(
    const float* __restrict__ x,   const float* __restrict__ remb,
    const float* __restrict__ w1g, const float* __restrict__ w2g,
    const float* __restrict__ w3g, const int* __restrict__ sel,
    const float* __restrict__ topw, const int* __restrict__ topi,
    const float* __restrict__ elog, float* __restrict__ out)
{
  extern __shared__ float sm[];
  float* xs   = sm + OFF_XS;
  float* w1s  = sm + OFF_W1S;     // 2 x (FC x LD1)
  float* w3s  = sm + OFF_W3S;     // 2 x (FC x LD1)
  float* h1b  = sm + OFF_H1;
  float* h3b  = sm + OFF_H3;
  float* hcb  = sm + OFF_HC;
  float* w2s  = sm + OFF_W2S;
  float* topwS= sm + OFF_TOPW;
  int*   topiS= (int*)(sm + OFF_TOPI);
  float* aew  = sm + OFF_AEW;

  const int tid  = threadIdx.x;
  const int lane = tid & 31;
  const int wv   = tid >> 5;
  const int half = lane >> 4;     // 0|1 : K-pair / M+8 selector in frag layouts
  const int lm   = lane & 15;
  const int64_t tok0 = (int64_t)blockIdx.x * MT;

  // ---- async-issue the x tile (32 x 768); first consumed after a wait+barrier ----
  for (int i = tid; i < MT * (CDIM / 4); i += 256) {
    const int r = i / (CDIM / 4), c4 = i % (CDIM / 4);
    cp16(&xs[r * LDX + c4 * 4], &x[(tok0 + r) * CDIM + c4 * 4]);
  }
  // ---- per-token meta ----
  if (tid < MT) {
    const int64_t t = tok0 + tid;
    topwS[tid * 2 + 0] = topw[t * 2 + 0];
    topwS[tid * 2 + 1] = topw[t * 2 + 1];
    topiS[tid * 2 + 0] = topi[t * 2 + 0];
    topiS[tid * 2 + 1] = topi[t * 2 + 1];
  }
  const int s0 = sel[0], s1 = sel[1];
  if (tid < 2 * MT) {
    const int k = tid >> 5, t = tid & 31;
    const int s = k ? s1 : s0;
    const int64_t tk = tok0 + t;
    const float l0 = elog[tk * NEXP + s * EPA];
    const float l1 = elog[tk * NEXP + s * EPA + 1];
    const float m = fmaxf(l0, l1);
    const float e0 = __expf(l0 - m), e1 = __expf(l1 - m);
    const float inv = 1.f / (e0 + e1);
    aew[(k * 2 + 0) * MT + t] = e0 * inv;
    aew[(k * 2 + 1) * MT + t] = e1 * inv;
  }
  __syncthreads();

  for (int k = 0; k < 2; ++k) {
    const int s = k ? s1 : s0;
    const float* w1b = w1g + (int64_t)(s * EPA) * FFN * CDIM;
    const float* w3b = w3g + (int64_t)(s * EPA) * FFN * CDIM;
    const float* w2b = w2g + (int64_t)(s * EPA) * CDIM * FFN;

    v8f acc[2][6];
#pragma unroll
    for (int mt = 0; mt < 2; ++mt)
#pragma unroll
      for (int j = 0; j < 6; ++j) acc[mt][j] = (v8f){};

    for (int e = 0; e < EPA; ++e) {
      const float* w1e = w1b + (int64_t)e * FFN * CDIM;
      const float* w3e = w3b + (int64_t)e * FFN * CDIM;
      const float* w2e = w2b + (int64_t)e * CDIM * FFN;

      for (int f0 = 0; f0 < FFN; f0 += FC) {
        __syncthreads();   // previous stage-2 readers (w2s/hcb) are done

        // async-issue w2 chunk (768 x 32) -- consumed a whole stage later
        for (int i = tid; i < CDIM * (FC / 4); i += 256) {
          const int r = i / (FC / 4), c4 = i % (FC / 4);
          cp16(&w2s[r * LDH + c4 * 4], &w2e[(int64_t)r * FFN + f0 + c4 * 4]);
        }
        // async-issue first w1/w3 chunk (kc=0) into ping buffer 0
        for (int i = tid; i < FC * (KC / 4); i += 256) {
          const int r = i / (KC / 4), c4 = i % (KC / 4);
          cp16(&w1s[r * LD1 + c4 * 4], &w1e[(int64_t)(f0 + r) * CDIM + c4 * 4]);
          cp16(&w3s[r * LD1 + c4 * 4], &w3e[(int64_t)(f0 + r) * CDIM + c4 * 4]);
        }

        // ---------------- stage 1: h1/h3 tiles over K=768, double-buffered ----------------
        const int tsel = wv & 3, mt1 = tsel & 1, nt1 = tsel >> 1;
        v8f hacc = (v8f){};
        int cur = 0;
        for (int kc = 0; kc < CDIM; kc += KC) {
          async_wait0();       // this wave's in-flight copies have landed in LDS
          __syncthreads();     // everyone's copies visible; pong readers done
          if (kc + KC < CDIM) {
            float* d1 = w1s + (cur ^ 1) * (FC * LD1);
            float* d3 = w3s + (cur ^ 1) * (FC * LD1);
            for (int i = tid; i < FC * (KC / 4); i += 256) {
              const int r = i / (KC / 4), c4 = i % (KC / 4);
              cp16(&d1[r * LD1 + c4 * 4],
                   &w1e[(int64_t)(f0 + r) * CDIM + kc + KC + c4 * 4]);
              cp16(&d3[r * LD1 + c4 * 4],
                   &w3e[(int64_t)(f0 + r) * CDIM + kc + KC + c4 * 4]);
            }
          }
          const float* bb = ((wv < 4) ? w1s : w3s) + cur * (FC * LD1);
#pragma unroll
          for (int kk = 0; kk < KC; kk += 4) {
            const int kb = kk + 2 * half;
            const v2f a = *(const v2f*)&xs[(mt1 * 16 + lm) * LDX + kc + kb];
            const v2f b = *(const v2f*)&bb[(nt1 * 16 + lm) * LD1 + kb];
            hacc = wmma4(a, b, hacc);
          }
          cur ^= 1;
        }
        {
          float* hb = (wv < 4) ? h1b : h3b;
#pragma unroll
          for (int r = 0; r < 8; ++r)
            hb[(mt1 * 16 + r + 8 * half) * LDH + nt1 * 16 + lm] = hacc[r];
        }
        __syncthreads();

        // ----- elementwise: h = ae[t,e] * silu(h1) * h3 -----
        for (int q = tid; q < MT * FC; q += 256) {
          const int t = q >> 5, f = q & 31;
          const float a = h1b[t * LDH + f];
          const float b = h3b[t * LDH + f];
          const float sg = a / (1.f + __expf(-a));
          hcb[t * LDH + f] = aew[(k * 2 + e) * MT + t] * sg * b;
        }
        __syncthreads();     // hcb ready; w2s arrived long ago (waited at kc=0)

        // ---------------- stage 2: out(32x768) += h(32x32) @ w2^T ----------------
#pragma unroll
        for (int kk = 0; kk < FC; kk += 4) {
          const int kb = kk + 2 * half;
          v2f bfr[6];
#pragma unroll
          for (int j = 0; j < 6; ++j) {
            const int n = (wv + 8 * j) * 16 + lm;      // output column c
            bfr[j] = *(const v2f*)&w2s[n * LDH + kb];
          }
#pragma unroll
          for (int mt = 0; mt < 2; ++mt) {
            const v2f a = *(const v2f*)&hcb[(mt * 16 + lm) * LDH + kb];
#pragma unroll
            for (int j = 0; j < 6; ++j) acc[mt][j] = wmma4(a, bfr[j], acc[mt][j]);
          }
        }
      } // f0
    }   // e

    // ---- epilogue for this k: scale by agent weight & role embedding ----
    float* op = out + tok0 * CDIM;
#pragma unroll
    for (int mt = 0; mt < 2; ++mt)
#pragma unroll
      for (int j = 0; j < 6; ++j) {
        const int c = (wv + 8 * j) * 16 + lm;
#pragma unroll
        for (int r = 0; r < 8; ++r) {
          const int t   = mt * 16 + r + 8 * half;     // C/D layout: M = r + 8*(lane>=16)
          const float aw = topwS[t * 2 + k];
          const int   id = topiS[t * 2 + k];
          const float em = remb[(int64_t)id * CDIM + c];
          const float v  = acc[mt][j][r] * aw * (1.f + 0.1f * em);
          const int64_t o = (int64_t)t * CDIM + c;
          if (k == 0) op[o] = v; else op[o] += v;
        }
      }
    __syncthreads();
  } // k
}

// ---------------------------------------------------------------------------
extern "C" void kernel_launch(void* const* d_in, const int* in_sizes, int n_in,
                              void* d_out, int out_size, void* d_ws, size_t ws_size,
                              hipStream_t stream) {
  const float* x    = (const float*)d_in[0];
  const float* agw  = (const float*)d_in[1];
  const float* egw  = (const float*)d_in[2];
  const float* remb = (const float*)d_in[3];
  const float* w1   = (const float*)d_in[4];
  const float* w2   = (const float*)d_in[5];
  const float* w3   = (const float*)d_in[6];
  float* out = (float*)d_out;

  char* ws = (char*)d_ws;
  int*   sel  = (int*)ws;                                   // 2 ints
  float* topw = (float*)(ws + 16);                          // NTOK*2 f32
  int*   topi = (int*)(ws + 16 + (size_t)NTOK * 2 * 4);     // NTOK*2 i32
  float* elog = (float*)(ws + 16 + (size_t)NTOK * 4 * 4);   // NTOK*20 f32

  moa_gate<<<NTOK / 8, 256, 0, stream>>>(x, agw, egw, sel, topw, topi, elog);

  const size_t smem = (size_t)SMEM_FLOATS * sizeof(float);
  moa_main<<<NTOK / MT, 256, smem, stream>>>(x, remb, w1, w2, w3,
                                             sel, topw, topi, elog, out);
}